// HeterogeneousRGCN_53077205844206
// MI455X (gfx1250) — compile-verified
//
#include <hip/hip_runtime.h>

#define D      128
#define ND     200000
#define NP     100000
#define NE     800000
#define ROWS   64      // destination rows per block in the WMMA kernel
#define LDA    130     // LDS row stride in floats (even -> 8B aligned v2f, spreads banks)

typedef __attribute__((ext_vector_type(2))) float v2f;
typedef __attribute__((ext_vector_type(8))) float v8f;

// ---------------------------------------------------------------------------
// Native fp32 global atomic add (no-return form -> STOREcnt, no CAS loop).
// ---------------------------------------------------------------------------
__device__ __forceinline__ void gatomic_add_f32(float* p, float v) {
  asm volatile("global_atomic_add_f32 %0, %1, off" :: "v"(p), "v"(v) : "memory");
}

// ---------------------------------------------------------------------------
// Zero fill (grid-stride).
// ---------------------------------------------------------------------------
__global__ void zero_f32(float* __restrict__ p, long n) {
  long i = (long)blockIdx.x * blockDim.x + threadIdx.x;
  long stride = (long)gridDim.x * blockDim.x;
  for (; i < n; i += stride) p[i] = 0.0f;
}

// ---------------------------------------------------------------------------
// Edge scatter: one wave32 per edge. Lane l handles 4 contiguous floats of the
// 128-float source row; gathers hit L2 (node tables fit in 192MB), atomic adds
// resolve in L2 atomic units. Lane 0 bumps the destination degree count.
// ---------------------------------------------------------------------------
__global__ __launch_bounds__(256)
void scatter_accum(const float* __restrict__ xsrc,
                   const int* __restrict__ esrc,
                   const int* __restrict__ edst,
                   float* __restrict__ agg,
                   float* __restrict__ cnt,
                   int nedges) {
  int wid  = (int)((blockIdx.x * blockDim.x + threadIdx.x) >> 5);
  int lane = threadIdx.x & 31;
  if (wid >= nedges) return;
  int s = esrc[wid];
  int d = edst[wid];
  const float4 v = *reinterpret_cast<const float4*>(xsrc + (size_t)s * D + lane * 4);
  float* p = agg + (size_t)d * D + lane * 4;
  gatomic_add_f32(p + 0, v.x);
  gatomic_add_f32(p + 1, v.y);
  gatomic_add_f32(p + 2, v.z);
  gatomic_add_f32(p + 3, v.w);
  if (lane == 0) gatomic_add_f32(cnt + d, 1.0f);
}

// ---------------------------------------------------------------------------
// Fused SAGE epilogue with fp32 WMMA:
//   out[r,:] (+)= (agg[r,:]/max(cnt[r],1)) @ Wl^T + b + xdst[r,:] @ Wr^T
// Block = 256 threads (8 waves). Block covers 64 dst rows x 128 cols.
// Wave w owns column tile n0 = 16*w and iterates 4 row tiles; per K-step of 4
// it issues 2x V_WMMA_F32_16X16X4_F32 per row tile, reusing the B fragments
// (weight columns) across the 4 row tiles.
//
// Fragment layouts (CDNA5 ISA 7.12.2, 32-bit):
//   A 16x4 : lane<16 -> row M=lane,    v0=K+0, v1=K+1 ; lane>=16 -> K+2,K+3
//   B 4x16 : lane<16 -> col N=lane,    v0=K+0, v1=K+1 ; lane>=16 -> K+2,K+3
//   C 16x16: vgpr v, lane -> (M = v + 8*(lane>>4), N = lane&15)
// B[k][n] = (W^T)[k][n] = W[n][k] -> each lane reads 2 consecutive floats of
// row n of W, which is contiguous in memory (v2f load, L2-resident weights).
// ---------------------------------------------------------------------------
__global__ __launch_bounds__(256)
void sage_wmma(const float* __restrict__ agg, const float* __restrict__ cnt,
               const float* __restrict__ xdst,
               const float* __restrict__ Wl, const float* __restrict__ bias,
               const float* __restrict__ Wr,
               float* __restrict__ out, int n, int relu, int accum) {
  __shared__ float Am[ROWS * LDA];  // mean tile
  __shared__ float Ax[ROWS * LDA];  // x_dst tile
  const int r0 = blockIdx.x * ROWS;

  // Cooperative staging: thread t -> row t>>2, column quarter t&3 (32 floats).
  {
    int row_l = threadIdx.x >> 2;
    int q     = threadIdx.x & 3;
    int r     = r0 + row_l;
    float inv = 0.0f;
    if (r < n) inv = 1.0f / fmaxf(cnt[r], 1.0f);
    #pragma unroll
    for (int i = 0; i < 8; i++) {
      int c0 = q * 32 + i * 4;
      float4 va = make_float4(0.f, 0.f, 0.f, 0.f);
      float4 vx = va;
      if (r < n) {
        va = *reinterpret_cast<const float4*>(agg  + (size_t)r * D + c0);
        vx = *reinterpret_cast<const float4*>(xdst + (size_t)r * D + c0);
      }
      float* pm = &Am[row_l * LDA + c0];
      pm[0] = va.x * inv; pm[1] = va.y * inv; pm[2] = va.z * inv; pm[3] = va.w * inv;
      float* px = &Ax[row_l * LDA + c0];
      px[0] = vx.x; px[1] = vx.y; px[2] = vx.z; px[3] = vx.w;
    }
  }
  __syncthreads();

  const int wave = threadIdx.x >> 5;      // column tile 0..7
  const int lane = threadIdx.x & 31;
  const int l15  = lane & 15;
  const int ncol = wave * 16 + l15;       // this lane's N column
  const int kb   = (lane >> 4) << 1;      // K sub-offset 0 or 2

  const float* wlrow = Wl + (size_t)ncol * D;
  const float* wrrow = Wr + (size_t)ncol * D;

  v8f acc[4] = {};

  for (int k0 = 0; k0 < D; k0 += 4) {
    const int kk = k0 + kb;
    const v2f bl = *reinterpret_cast<const v2f*>(wlrow + kk);
    const v2f br = *reinterpret_cast<const v2f*>(wrrow + kk);
    #pragma unroll
    for (int rt = 0; rt < 4; rt++) {
      const v2f am = *reinterpret_cast<const v2f*>(&Am[(rt * 16 + l15) * LDA + kk]);
      const v2f ax = *reinterpret_cast<const v2f*>(&Ax[(rt * 16 + l15) * LDA + kk]);
      acc[rt] = __builtin_amdgcn_wmma_f32_16x16x4_f32(
          false, am, false, bl, (short)0, acc[rt], false, false);
      acc[rt] = __builtin_amdgcn_wmma_f32_16x16x4_f32(
          false, ax, false, br, (short)0, acc[rt], false, false);
    }
  }

  const float bia = bias[ncol];
  const int mhi = (lane >> 4) << 3;
  #pragma unroll
  for (int rt = 0; rt < 4; rt++) {
    #pragma unroll
    for (int v = 0; v < 8; v++) {
      int r = r0 + rt * 16 + mhi + v;
      if (r < n) {
        size_t idx = (size_t)r * D + ncol;
        float val = acc[rt][v] + bia;
        if (accum) val += out[idx];
        if (relu)  val = fmaxf(val, 0.0f);
        out[idx] = val;
      }
    }
  }
}

// ---------------------------------------------------------------------------
// Host orchestration.
// ---------------------------------------------------------------------------
extern "C" void kernel_launch(void* const* d_in, const int* in_sizes, int n_in,
                              void* d_out, int out_size, void* d_ws, size_t ws_size,
                              hipStream_t stream) {
  (void)in_sizes; (void)n_in; (void)out_size; (void)ws_size;

  const float* xd0 = (const float*)d_in[0];
  const float* xp0 = (const float*)d_in[1];
  const float* W[18];
  for (int i = 0; i < 18; i++) W[i] = (const float*)d_in[2 + i];
  // W[0..2]=L0 dp (Wl,b,Wr)  W[3..5]=L0 pd  W[6..8]=L0 dd
  // W[9..11]=L1 dp           W[12..14]=L1 pd W[15..17]=L1 dd
  const int* e_dp_s = (const int*)d_in[20]; const int* e_dp_d = (const int*)d_in[21];
  const int* e_pd_s = (const int*)d_in[22]; const int* e_pd_d = (const int*)d_in[23];
  const int* e_dd_s = (const int*)d_in[24]; const int* e_dd_d = (const int*)d_in[25];

  float* out_d = (float*)d_out;                      // [ND, D]
  float* out_p = out_d + (size_t)ND * D;             // [NP, D]

  float* agg = (float*)d_ws;                         // ND*D floats (max)
  float* cnt = agg + (size_t)ND * D;                 // ND floats
  float* xd1 = cnt + ND;                             // ND*D floats
  float* xp1 = xd1 + (size_t)ND * D;                 // NP*D floats

  const int scatter_blocks = (NE + 7) / 8;           // 8 waves of 32 per block

  auto run_rel = [&](const float* xsrc, const float* xdst,
                     const int* es, const int* ed, int ndst,
                     const float* Wl, const float* b, const float* Wr,
                     float* o, int relu, int accum) {
    zero_f32<<<2048, 256, 0, stream>>>(agg, (long)ndst * D);
    zero_f32<<<64,   256, 0, stream>>>(cnt, (long)ndst);
    scatter_accum<<<scatter_blocks, 256, 0, stream>>>(xsrc, es, ed, agg, cnt, NE);
    sage_wmma<<<(ndst + ROWS - 1) / ROWS, 256, 0, stream>>>(
        agg, cnt, xdst, Wl, b, Wr, o, ndst, relu, accum);
  };

  // ---- Layer 0: inputs (xd0, xp0) -> (xd1, xp1), ReLU applied ----
  // new_p = sage_dp(xd -> prot), ReLU
  run_rel(xd0, xp0, e_dp_s, e_dp_d, NP, W[0], W[1], W[2], xp1, /*relu=*/1, /*accum=*/0);
  // new_d = sage_pd(xp -> drug) + sage_dd(xd -> drug), ReLU on the sum
  run_rel(xp0, xd0, e_pd_s, e_pd_d, ND, W[3], W[4], W[5], xd1, 0, 0);
  run_rel(xd0, xd0, e_dd_s, e_dd_d, ND, W[6], W[7], W[8], xd1, 1, 1);

  // ---- Layer 1: inputs (xd1, xp1) -> d_out, no activation ----
  run_rel(xd1, xp1, e_dp_s, e_dp_d, NP, W[9],  W[10], W[11], out_p, 0, 0);
  run_rel(xp1, xd1, e_pd_s, e_pd_d, ND, W[12], W[13], W[14], out_d, 0, 0);
  run_rel(xd1, xd1, e_dd_s, e_dd_d, ND, W[15], W[16], W[17], out_d, 0, 1);
}